// YOLOLoss_83691732730327
// MI455X (gfx1250) — compile-verified
//
#include <hip/hip_runtime.h>
#include <math.h>

#define EPSL 1e-3f

typedef float v2f __attribute__((ext_vector_type(2)));
typedef float v8f __attribute__((ext_vector_type(8)));
typedef unsigned int u32x4 __attribute__((ext_vector_type(4)));
typedef int i32x4 __attribute__((ext_vector_type(4)));
typedef int i32x8 __attribute__((ext_vector_type(8)));

// ---------------- workspace layout (32-bit word offsets) ----------------
enum {
  OFF_VALID = 0,        // int[512]
  OFF_BAI   = 512,      // int[512]  best anchor index
  OFF_BSI   = 1024,     // int[512]  best scale index
  OFF_CLS   = 1536,     // int[512]  class index
  OFF_XI    = 2048,     // int[3][512] cell x per scale
  OFF_YI    = 3584,     // int[3][512] cell y per scale
  OFF_TOBJ  = 5120,     // float[3][3][512] target_obj (scale, anchor, tgt)
  OFF_MM    = 9728,     // int[3][3][512]   best|overlap mask
  OFF_BOXB  = 14336,    // float[4][512]    box targets
  OFF_OWIN  = 16384,    // int[3][512]      obj/cls scatter winner
  OFF_BWIN  = 17920,    // int[3][512]      box scatter winner
  OFF_ACC   = 19456     // float[8]: 0 objS 1 objC 2 clsS 3 clsC 4 boxS 5 boxC
};

__device__ __forceinline__ float softplusf_(float x) {
  return fmaxf(x, 0.f) + log1pf(expf(-fabsf(x)));
}
__device__ __forceinline__ float logitf_(float x) {
  if (x == 0.f) x = EPSL;
  if (x == 1.f) x = 1.f - EPSL;
  return logf(x / (1.f - x));
}
__device__ __forceinline__ float safelogf_(float x) {
  return logf(fmaxf(x, EPSL));
}

// ------------- kernel 1: per-target prior matching (N = 512) -------------
__global__ void k_targets(const float* __restrict__ anchors,
                          const float* __restrict__ strides,
                          const float* __restrict__ targets,
                          float* __restrict__ F, int* __restrict__ I) {
  int j = blockIdx.x * blockDim.x + threadIdx.x;
  if (j >= 512) return;
  if (j == 0) {  // init accumulators (runs before k_obj/k_corr on stream)
    for (int q = 0; q < 8; ++q) F[OFF_ACC + q] = 0.f;
    F[OFF_ACC + 1] = 403200.f;  // baseline obj count: sum b*a*H*W
  }
  const float* t = targets + (size_t)j * 5;
  float t0 = t[0], t1 = t[1], t2 = t[2], t3 = t[3], t4 = t[4];
  bool mask = !(t0 == -1.f && t1 == -1.f && t2 == -1.f && t3 == -1.f && t4 == -1.f);

  float iou[3][3], offx[3], offy[3], whw[3], whh[3];
  for (int i = 0; i < 3; ++i) {
    float st = strides[i];
    float tbx = mask ? t0 / st : t0;
    float tby = mask ? t1 / st : t1;
    float tbw = mask ? t2 / st : t2;
    float tbh = mask ? t3 / st : t3;
    int cx = (int)tbx, cy = (int)tby;            // astype(int32) truncation
    float ox = mask ? tbx - (float)cx : tbx;
    float oy = mask ? tby - (float)cy : tby;
    offx[i] = ox; offy[i] = oy; whw[i] = tbw; whh[i] = tbh;
    I[OFF_XI + i * 512 + j] = cx;
    I[OFF_YI + i * 512 + j] = cy;
    float a1x0 = ox - tbw * 0.5f, a1y0 = oy - tbh * 0.5f;
    float a1x1 = ox + tbw * 0.5f, a1y1 = oy + tbh * 0.5f;
    float area1 = (a1x1 - a1x0) * (a1y1 - a1y0);
    for (int k = 0; k < 3; ++k) {
      float aw = anchors[(i * 3 + k) * 2 + 0] / st;
      float ah = anchors[(i * 3 + k) * 2 + 1] / st;
      float a2x0 = 0.5f - aw * 0.5f, a2y0 = 0.5f - ah * 0.5f;
      float a2x1 = 0.5f + aw * 0.5f, a2y1 = 0.5f + ah * 0.5f;
      float x0 = fmaxf(a1x0, a2x0), y0 = fmaxf(a1y0, a2y0);
      float x1 = fminf(a1x1, a2x1), y1 = fminf(a1y1, a2y1);
      float inter = (x1 - x0) * (y1 - y0) * ((x0 < x1 && y0 < y1) ? 1.f : 0.f);
      float area2 = aw * ah;
      iou[i][k] = inter / (area1 + area2 - inter);
    }
  }
  // argmax over flattened (s*a), first-max-wins like jnp.argmax
  int bf = 0; float bv = iou[0][0];
  for (int f = 1; f < 9; ++f) {
    float v = iou[f / 3][f % 3];
    if (v > bv) { bv = v; bf = f; }
  }
  int bs = bf / 3, ba = bf % 3;
  I[OFF_VALID + j] = mask ? 1 : 0;
  I[OFF_BSI + j] = bs;
  I[OFF_BAI + j] = ba;
  int ci = (int)t4; ci %= 80; if (ci < 0) ci += 80;   // jnp.mod semantics
  I[OFF_CLS + j] = ci;
  for (int i = 0; i < 3; ++i)
    for (int k = 0; k < 3; ++k) {
      bool best = (i == bs && k == ba);
      bool over = (iou[i][k] > 0.5f) && !best;
      F[OFF_TOBJ + (i * 3 + k) * 512 + j] = best ? 1.f : (over ? -1.f : 0.f);
      I[OFF_MM + (i * 3 + k) * 512 + j] = (best || over) ? 1 : 0;
    }
  // box targets at the best (scale, anchor)
  float aw = anchors[(bs * 3 + ba) * 2 + 0] / strides[bs];
  float ah = anchors[(bs * 3 + ba) * 2 + 1] / strides[bs];
  F[OFF_BOXB + 0 * 512 + j] = logitf_(offx[bs]);
  F[OFF_BOXB + 1 * 512 + j] = logitf_(offy[bs]);
  F[OFF_BOXB + 2 * 512 + j] = safelogf_(whw[bs] / aw);
  F[OFF_BOXB + 3 * 512 + j] = safelogf_(whh[bs] / ah);
}

// ---- kernel 2: scatter collision resolution (last writer wins, per scale) ----
__global__ void k_winners(int* __restrict__ I) {
  int idx = blockIdx.x * blockDim.x + threadIdx.x;
  if (idx >= 3 * 512) return;
  int i = idx / 512, j = idx % 512;
  bool v = I[OFF_VALID + j] != 0;
  int x = I[OFF_XI + i * 512 + j], y = I[OFF_YI + i * 512 + j];
  int ai = I[OFF_BAI + j];
  bool ow = v;
  bool bw = v && (I[OFF_BSI + j] == i);
  for (int j2 = j + 1; j2 < 512; ++j2) {
    if (!I[OFF_VALID + j2]) continue;
    if ((j2 >> 5) != (j >> 5)) continue;          // same batch (n=32)
    if (I[OFF_XI + i * 512 + j2] == x && I[OFF_YI + i * 512 + j2] == y) {
      ow = false;                                  // obj/cls row overwritten
      if (I[OFF_BSI + j2] == i && I[OFF_BAI + j2] == ai) bw = false;
    }
  }
  I[OFF_OWIN + i * 512 + j] = ow ? 1 : 0;
  I[OFF_BWIN + i * 512 + j] = bw ? 1 : 0;
}

// ----- kernel 3: dense obj baseline  sum(0.5*softplus(po)) over 403200 ------
// The channel-4 column (stride 85 floats = 340 B) is gathered by the Tensor
// Data Mover into contiguous LDS tiles (tile_dim0 = 1 elem, tile_dim1 = 1280
// rows, tensor_dim0_stride = 85), double-buffered against TENSORcnt, so the
// waves only ever do dense ds_load + softplus.  315 chunks of 1280 cells
// exactly cover 307200 + 76800 + 19200 positions (no OOB zero-fill).
#define CHUNK 1280
#define NCHUNK 315   // 240 + 60 + 15

__device__ __forceinline__ void tdm_issue(const float* base, int cellStart,
                                          unsigned ldsByteAddr) {
  unsigned long long ga =
      (unsigned long long)(uintptr_t)base +
      ((unsigned long long)cellStart * 85ull + 4ull) * 4ull;
  u32x4 g0;
  g0.x = 1u;                                        // count=1, load descriptor
  g0.y = ldsByteAddr;                               // lds_addr (bytes)
  g0.z = (unsigned)(ga & 0xFFFFFFFFull);            // global_addr[31:0]
  g0.w = (unsigned)((ga >> 32) & 0x01FFFFFFull) | (2u << 30);  // addr[56:32]|type=2
  i32x8 g1;
  g1[0] = (int)(2u << 16);                          // data_size=4B, no multicast
  g1[1] = (int)(85u << 16);                         // tensor_dim0=85 (bits 79:48 lo)
  g1[2] = (int)((unsigned)CHUNK << 16);             // td0 hi=0 | tensor_dim1 lo
  g1[3] = (int)(1u << 16);                          // td1 hi=0 | tile_dim0=1
  g1[4] = CHUNK;                                    // tile_dim1=1280, tile_dim2=0
  g1[5] = 85;                                       // tensor_dim0_stride lo
  g1[6] = 0;                                        // stride0 hi | stride1 lo
  g1[7] = 0;                                        // stride1 hi
  i32x4 gz = {0, 0, 0, 0};
#if defined(__clang_major__) && (__clang_major__ >= 23)
  i32x8 gz8 = {0, 0, 0, 0, 0, 0, 0, 0};
  __builtin_amdgcn_tensor_load_to_lds(g0, g1, gz, gz, gz8, 0);
#else
  __builtin_amdgcn_tensor_load_to_lds(g0, g1, gz, gz, 0);
#endif
}

__global__ void k_obj(const float* __restrict__ inf0, const float* __restrict__ inf1,
                      const float* __restrict__ inf2, float* __restrict__ F) {
  __shared__ float tile[2][CHUNK];
  const bool wave0 = (threadIdx.x >> 5) == 0;
  const unsigned ldsBase = (unsigned)(uintptr_t)&tile[0][0];

  // chunk -> (tensor, first cell)
  auto chunk_base = [&](int g, const float*& base, int& cell) {
    if (g < 240)      { base = inf0; cell = g * CHUNK; }
    else if (g < 300) { base = inf1; cell = (g - 240) * CHUNK; }
    else              { base = inf2; cell = (g - 300) * CHUNK; }
  };

  float acc0 = 0.f, acc1 = 0.f;
  int g0c = blockIdx.x;
  int stride = gridDim.x;

  if (wave0 && g0c < NCHUNK) {
    const float* base; int cell;
    chunk_base(g0c, base, cell);
    tdm_issue(base, cell, ldsBase);                 // prologue -> buffer 0
  }
  int i = 0;
  for (int g = g0c; g < NCHUNK; g += stride, ++i) {
    int gn = g + stride;
    if (wave0) {
      if (gn < NCHUNK) {                            // issue next into other buffer
        const float* base; int cell;
        chunk_base(gn, base, cell);
        tdm_issue(base, cell, ldsBase + ((unsigned)((i + 1) & 1)) * CHUNK * 4u);
        __builtin_amdgcn_s_wait_tensorcnt((short)1);  // current chunk landed
      } else {
        __builtin_amdgcn_s_wait_tensorcnt((short)0);
      }
    }
    __syncthreads();                                // publish LDS tile to all waves
    const float* tp = tile[i & 1];
    #pragma unroll
    for (int k = 0; k < CHUNK / 256; ++k) {
      float v = softplusf_(tp[threadIdx.x + k * 256]);
      if (k & 1) acc1 += v; else acc0 += v;
    }
    __syncthreads();                                // done reading before reuse
  }

  // Cross-lane reduction of 64 wave partials via one f32 WMMA:
  // D = A(16x4) * ones(4x16) + 0  ->  every D row = its row-sum.
  v2f a; a.x = acc0 * 0.5f; a.y = acc1 * 0.5f;      // fold BCE weight 0.5
  v2f bones; bones.x = 1.f; bones.y = 1.f;
  v8f c = {};
  c = __builtin_amdgcn_wmma_f32_16x16x4_f32(
      /*neg_a=*/false, a, /*neg_b=*/false, bones,
      /*c_mod=*/(short)0, c, /*reuse_a=*/false, /*reuse_b=*/false);
  float part = c[0] + c[1] + c[2] + c[3] + c[4] + c[5] + c[6] + c[7];
  // lanes 0-15 hold rows 0-7, lanes 16-31 hold rows 8-15 -> add both halves
  part += __shfl_xor(part, 16, 32);
  if ((threadIdx.x & 31) == 0) atomicAdd(&F[OFF_ACC + 0], part);
}

// -------- kernel 4: sparse corrections (obj overrides, cls BCE, box MSE) ------
__global__ void k_corr(const float* __restrict__ inf0, const float* __restrict__ inf1,
                       const float* __restrict__ inf2,
                       float* __restrict__ F, const int* __restrict__ I) {
  int idx = blockIdx.x * blockDim.x + threadIdx.x;
  if (idx >= 3 * 512) return;
  int i = idx / 512, j = idx % 512;
  const float* inf = (i == 0) ? inf0 : (i == 1) ? inf1 : inf2;
  int H = (i == 0) ? 80 : (i == 1) ? 40 : 20;   // W == H
  int b = j >> 5;
  int x = I[OFF_XI + i * 512 + j], y = I[OFF_YI + i * 512 + j];

  if (I[OFF_OWIN + i * 512 + j]) {
    float dObj = 0.f, dCnt = 0.f, dCls = 0.f, dClsCnt = 0.f;
    int ci = I[OFF_CLS + j];
    for (int k = 0; k < 3; ++k) {
      size_t cell = ((((size_t)(b * 3 + k)) * H + y) * H + x) * 85;
      __builtin_prefetch(&inf[cell + 5], 0, 1);
      float po = inf[cell + 4];
      float spv = softplusf_(po);
      float go = F[OFF_TOBJ + (i * 3 + k) * 512 + j];
      if (go > 0.5f)       dObj += (spv - po) * 2.f - spv * 0.5f; // replace baseline
      else if (go < -0.5f) { dObj -= spv * 0.5f; dCnt -= 1.f; }   // ignored cell
      if (I[OFF_MM + (i * 3 + k) * 512 + j]) {                     // cls BCE row
        float cs = 0.f;
        for (int c = 0; c < 80; ++c) cs += softplusf_(inf[cell + 5 + c]);
        cs -= inf[cell + 5 + ci];                                  // - x*z, z one-hot
        dCls += cs; dClsCnt += 1.f;
      }
    }
    atomicAdd(&F[OFF_ACC + 0], dObj);
    atomicAdd(&F[OFF_ACC + 1], dCnt);
    atomicAdd(&F[OFF_ACC + 2], dCls);
    atomicAdd(&F[OFF_ACC + 3], dClsCnt);
  }
  if (I[OFF_BWIN + i * 512 + j]) {
    int ai = I[OFF_BAI + j];
    size_t cell = ((((size_t)(b * 3 + ai)) * H + y) * H + x) * 85;
    float bsum = 0.f;
    for (int c = 0; c < 4; ++c) {
      float d = F[OFF_BOXB + c * 512 + j] - inf[cell + c];
      bsum += d * d;
    }
    atomicAdd(&F[OFF_ACC + 4], bsum);
    atomicAdd(&F[OFF_ACC + 5], 1.f);
  }
}

// ------------------------- kernel 5: final combine ---------------------------
__global__ void k_final(const float* __restrict__ F, float* __restrict__ out) {
  float box = 5.f * F[OFF_ACC + 4] / F[OFF_ACC + 5];
  float obj = F[OFF_ACC + 0] / F[OFF_ACC + 1];
  float cls = F[OFF_ACC + 2] / (F[OFF_ACC + 3] * 80.f);
  out[0] = box + obj + cls;
}

extern "C" void kernel_launch(void* const* d_in, const int* in_sizes, int n_in,
                              void* d_out, int out_size, void* d_ws, size_t ws_size,
                              hipStream_t stream) {
  const float* anchors = (const float*)d_in[0];
  const float* strides = (const float*)d_in[1];
  const float* inf0    = (const float*)d_in[2];
  const float* inf1    = (const float*)d_in[3];
  const float* inf2    = (const float*)d_in[4];
  const float* targets = (const float*)d_in[5];
  float* F = (float*)d_ws;
  int*   I = (int*)d_ws;
  float* out = (float*)d_out;

  k_targets<<<2, 256, 0, stream>>>(anchors, strides, targets, F, I);
  k_winners<<<6, 256, 0, stream>>>(I);
  k_obj<<<105, 256, 0, stream>>>(inf0, inf1, inf2, F);     // 315 chunks / 105 blocks
  k_corr<<<6, 256, 0, stream>>>(inf0, inf1, inf2, F, I);
  k_final<<<1, 1, 0, stream>>>(F, out);
}